// Direction_61495341744914
// MI455X (gfx1250) — compile-verified
//
#include <hip/hip_runtime.h>
#include <hip/hip_bf16.h>

typedef __attribute__((ext_vector_type(2))) float v2f;
typedef __attribute__((ext_vector_type(8))) float v8f;

#define BATCH 262144
#define MDIM  20
#define FEAT  512

// -------- block-wide sum over 512 threads (16 wave32 waves) --------
__device__ __forceinline__ float blk_sum(float x, float* red, int lane, int wave) {
    #pragma unroll
    for (int off = 16; off > 0; off >>= 1)
        x += __shfl_xor(x, off, 32);
    __syncthreads();                 // protect red[] from previous iteration's readers
    if (lane == 0) red[wave] = x;
    __syncthreads();
    float s = 0.f;
    #pragma unroll
    for (int w = 0; w < 16; ++w) s += red[w];
    return s;
}

// -------- QR of W (512x20) via modified Gram-Schmidt, one thread per row ------
// Writes Q pre-swizzled for WMMA B fragments:
//   P (as v2f): P[kp*FEAT + n] = { Q[n][2kp], Q[n][2kp+1] },  kp = 0..9
__global__ __launch_bounds__(512)
void qr_mgs_kernel(const float* __restrict__ W, float* __restrict__ Pout) {
    __shared__ float red[16];
    __shared__ float piv;
    const int t    = threadIdx.x;    // row of W / row of Q, 0..511
    const int lane = t & 31;
    const int wave = t >> 5;

    float v[MDIM];
    #pragma unroll
    for (int m = 0; m < MDIM; ++m) v[m] = W[t * MDIM + m] + 1e-8f;

    #pragma unroll
    for (int j = 0; j < MDIM; ++j) {
        // classical-in-order MGS projections against already-formed columns
        #pragma unroll
        for (int i = 0; i < j; ++i) {
            float d = blk_sum(v[i] * v[j], red, lane, wave);
            v[j] -= d * v[i];
        }
        float nrm2 = blk_sum(v[j] * v[j], red, lane, wave);
        // LAPACK Householder sign convention: beta = -sign(pivot)*||x||
        __syncthreads();
        if (t == j) piv = v[j];
        __syncthreads();
        float inv = 1.0f / sqrtf(nrm2);
        if (piv >= 0.f) inv = -inv;
        v[j] *= inv;                 // column j is now q_j
    }

    v2f* Pp = (v2f*)Pout;
    #pragma unroll
    for (int kp = 0; kp < MDIM / 2; ++kp) {
        v2f e; e.x = v[2 * kp]; e.y = v[2 * kp + 1];
        Pp[kp * FEAT + t] = e;
    }
}

// -------- out = input @ Q^T  via V_WMMA_F32_16X16X4_F32 --------
// Wave-level: each wave owns a 16-row batch tile, loops over all 32 N-tiles.
// A fragment (16x4 f32): lanes 0-15 row M=l {K, K+1}, lanes 16-31 row M=l-16 {K+2, K+3}.
// B fragment mirrors it; the packed-Q layout makes it one b64 load per lane.
// C layout: VGPR r -> M=r (lanes 0-15) / M=r+8 (lanes 16-31), N = lane%16.
__global__ __launch_bounds__(256)
void wmma_gemm_kernel(const float* __restrict__ A,
                      const float* __restrict__ P,
                      float* __restrict__ Out) {
    const int lane = threadIdx.x & 31;
    const int wave = threadIdx.x >> 5;   // 0..7
    const int half = lane >> 4;          // 0 or 1
    const int ln   = lane & 15;

    const size_t tile = (size_t)blockIdx.x * 8 + wave;   // 16-row tile index
    const size_t row  = tile * 16 + ln;

    // A fragments for the 5 K-blocks (K = 20 = 5 * 4), 8B aligned loads
    const float* arow = A + row * MDIM;
    v2f a[5];
    #pragma unroll
    for (int kb = 0; kb < 5; ++kb)
        a[kb] = *(const v2f*)(arow + 4 * kb + 2 * half);

    const v2f* Bp  = (const v2f*)P;      // 40 KB total, L2-resident
    float* orow = Out + tile * 16 * FEAT;

    for (int nt = 0; nt < FEAT / 16; ++nt) {
        v8f c = {0.f, 0.f, 0.f, 0.f, 0.f, 0.f, 0.f, 0.f};
        #pragma unroll
        for (int kb = 0; kb < 5; ++kb) {
            v2f b = Bp[(size_t)(2 * kb + half) * FEAT + nt * 16 + ln];
            c = __builtin_amdgcn_wmma_f32_16x16x4_f32(
                    /*neg_a=*/false, a[kb],
                    /*neg_b=*/false, b,
                    /*c_mod=*/(short)0, c,
                    /*reuse_a=*/false, /*reuse_b=*/false);
        }
        #pragma unroll
        for (int r = 0; r < 8; ++r)
            orow[(size_t)(r + 8 * half) * FEAT + nt * 16 + ln] = c[r];
    }
}

extern "C" void kernel_launch(void* const* d_in, const int* in_sizes, int n_in,
                              void* d_out, int out_size, void* d_ws, size_t ws_size,
                              hipStream_t stream) {
    const float* input  = (const float*)d_in[0];   // (262144, 20) f32
    const float* weight = (const float*)d_in[1];   // (512, 20) f32
    float* out = (float*)d_out;                    // (262144, 512) f32
    float* P   = (float*)d_ws;                     // 512*20 floats = 40 KB packed Q

    qr_mgs_kernel<<<1, 512, 0, stream>>>(weight, P);

    const int blocks = BATCH / (16 * 8);           // 2048
    wmma_gemm_kernel<<<blocks, 256, 0, stream>>>(input, P, out);
}